// ParticleFilter_10831907520858
// MI455X (gfx1250) — compile-verified
//
#include <hip/hip_runtime.h>
#include <hip/hip_bf16.h>
#include <math.h>

typedef __attribute__((ext_vector_type(2)))  float    v2f;
typedef __attribute__((ext_vector_type(8)))  float    v8f;

#define PN        1048576
#define NSENS     64
#define MAP_SIZE  20.0f
#define DDIR_K    0.06283185307179586f   /* 2*pi*THETA_NOISE */
#define SPD_K     0.02f
#define INV_2VAR  5.0f                   /* 1/(2*(0.3^2+0.1^2)) */
#define LOG_UNIF  -13.862943611198906f   /* -log(N) */
#define BIGF      3.0e38f

// ---------------------------------------------------------------------------
// K1: predict + WMMA pairwise distances + two-smallest + log-weight + block max
//     block = 256 threads = 8 waves; each wave owns 32 particles.
//     A (sensor) matrices are pre-packed in LDS in the exact per-lane WMMA
//     layout so the inner loop issues one unconditional ds_load_b64 per tile
//     (no divergent-EXEC conditional LDS reads).
// ---------------------------------------------------------------------------
__global__ __launch_bounds__(256) void k_predict_weight(
    const float* __restrict__ x, const float* __restrict__ p,
    const float* __restrict__ sensors, const float* __restrict__ z,
    const float* __restrict__ nth, const float* __restrict__ nsp,
    const float* __restrict__ delta,
    float* __restrict__ xnew, float* __restrict__ w, float* __restrict__ pmax)
{
    // aPack[t][lane][0..1]: A-tile operand for sensor-tile t as seen by `lane`.
    //   lanes 0-15  (K0,K1): (-2*sx, -2*sy)      for sensor s = 16t + lane
    //   lanes 16-31 (K2,K3): (sx^2+sy^2, 1.0f)   for sensor s = 16t + lane-16
    __shared__ float aPack[4 * 32 * 2];
    __shared__ float red[256];

    const int t    = threadIdx.x;
    const int lane = t & 31;
    const int i    = blockIdx.x * 256 + t;     // my particle

    {
        // 256 threads build exactly 256 packed floats.
        int tile = t >> 6;            // 0..3
        int rem  = t & 63;
        int l    = rem >> 1;          // lane 0..31
        int comp = rem & 1;           // 0..1
        int s    = tile * 16 + (l & 15);
        float sx = sensors[2 * s], sy = sensors[2 * s + 1];
        float v;
        if (l < 16) v = comp ? (-2.0f * sy) : (-2.0f * sx);
        else        v = comp ? 1.0f : (sx * sx + sy * sy);
        aPack[t] = v;
    }
    __syncthreads();

    // ---- predict (motion) ----
    float px = x[3 * i + 0], py = x[3 * i + 1], th = x[3 * i + 2];
    float dd  = DDIR_K * nth[i];
    float dir = th + dd;
    float dv  = SPD_K * nsp[i];
    px = px + delta[0] + dv * __cosf(dir);
    py = py + delta[1] + dv * __sinf(dir);
    float thn = th + delta[2] + dd;
    px = fminf(fmaxf(px, 0.0f), MAP_SIZE);
    py = fminf(fmaxf(py, 0.0f), MAP_SIZE);
    xnew[3 * i + 0] = px;
    xnew[3 * i + 1] = py;
    xnew[3 * i + 2] = thn;
    float pp = px * px + py * py;

    // ---- distances via WMMA: D[s,n] = ||p_n - s_s||^2 (rank-4 GEMM) ----
    const int half = lane >> 4;        // 0: lanes 0-15, 1: lanes 16-31
    const int l16  = lane & 15;
    const v2f* aP = (const v2f*)aPack; // aP[t*32 + lane]

    float b1_0 = BIGF, b2_0 = BIGF, b1_1 = BIGF, b2_1 = BIGF;

#pragma unroll
    for (int j = 0; j < 2; ++j) {                 // particle tile (16 particles)
        int srcl = (j << 4) + l16;
        float bpx = __shfl(px, srcl, 32);
        float bpy = __shfl(py, srcl, 32);
        float bpp = __shfl(pp, srcl, 32);

        // B (4x16 f32): lanes 0-15 hold K0,K1 = (px,py);
        //               lanes 16-31 hold K2,K3 = (1, px^2+py^2)
        v2f b;
        b.x = half ? 1.0f : bpx;
        b.y = half ? bpp  : bpy;

        float a1 = BIGF, a2 = BIGF;               // two smallest seen by this lane
#pragma unroll
        for (int tt = 0; tt < 4; ++tt) {          // sensor tile (16 sensors)
            v2f a = aP[tt * 32 + lane];           // one ds_load_b64, uniform EXEC
            v8f c = {};
            c = __builtin_amdgcn_wmma_f32_16x16x4_f32(
                    false, a, false, b, (short)0, c, false, false);
#pragma unroll
            for (int v = 0; v < 8; ++v) {
                float d = c[v];
                float nm2 = fminf(a2, fmaxf(a1, d));
                a1 = fminf(a1, d);
                a2 = nm2;
            }
        }
        // merge this lane's 32 sensors with partner lane (lane ^ 16) -> all 64
        float o1 = __shfl_xor(a1, 16, 32);
        float o2 = __shfl_xor(a2, 16, 32);
        float lo  = fminf(a1, o1);
        float hi  = fmaxf(a1, o1);
        float sec = fminf(hi, fminf(a2, o2));
        if (j == 0) { b1_0 = lo; b2_0 = sec; } else { b1_1 = lo; b2_1 = sec; }
    }

    // lane l's own particle lives in tile (l>>4), column (l&15)
    float m1 = half ? b1_1 : b1_0;
    float m2 = half ? b2_1 : b2_0;
    float q1 = sqrtf(fmaxf(m1, 0.0f));
    float q2 = sqrtf(fmaxf(m2, 0.0f));
    float df0 = q1 - z[0];
    float df1 = q2 - z[1];
    float wv  = p[i] - (df0 * df0 + df1 * df1) * INV_2VAR;
    w[i] = wv;

    // ---- block max reduction ----
    red[t] = wv;
    __syncthreads();
    for (int s = 128; s > 0; s >>= 1) {
        if (t < s) red[t] = fmaxf(red[t], red[t + s]);
        __syncthreads();
    }
    if (t == 0) pmax[blockIdx.x] = red[0];
}

// ---------------------------------------------------------------------------
// K2: global max over 4096 block partials (single block)
// ---------------------------------------------------------------------------
__global__ __launch_bounds__(1024) void k_maxreduce(
    const float* __restrict__ pmax, float* __restrict__ scal)
{
    __shared__ float red[1024];
    int t = threadIdx.x;
    float m = -BIGF;
    for (int k = t; k < 4096; k += 1024) m = fmaxf(m, pmax[k]);
    red[t] = m;
    __syncthreads();
    for (int s = 512; s > 0; s >>= 1) {
        if (t < s) red[t] = fmaxf(red[t], red[t + s]);
        __syncthreads();
    }
    if (t == 0) scal[0] = red[0];
}

// ---------------------------------------------------------------------------
// K3: per-block (1024 elems) sum(e) and sum(e^2) with e = exp(w - gmax)
// ---------------------------------------------------------------------------
__global__ __launch_bounds__(256) void k_sums(
    const float* __restrict__ w, const float* __restrict__ scal,
    float* __restrict__ bSum, float* __restrict__ bSum2)
{
    __shared__ float rs[256], rs2[256];
    int t = threadIdx.x, b = blockIdx.x;
    float gmax = scal[0];
    float4 v = ((const float4*)(w + (size_t)b * 1024))[t];
    float e0 = __expf(v.x - gmax), e1 = __expf(v.y - gmax);
    float e2 = __expf(v.z - gmax), e3 = __expf(v.w - gmax);
    rs[t]  = (e0 + e1) + (e2 + e3);
    rs2[t] = (e0 * e0 + e1 * e1) + (e2 * e2 + e3 * e3);
    __syncthreads();
    for (int s = 128; s > 0; s >>= 1) {
        if (t < s) { rs[t] += rs[t + s]; rs2[t] += rs2[t + s]; }
        __syncthreads();
    }
    if (t == 0) { bSum[b] = rs[0]; bSum2[b] = rs2[0]; }
}

// ---------------------------------------------------------------------------
// K4: single-block scan of the 1024 block sums -> per-block offsets, Z, logZ,
//     N_eff decision
// ---------------------------------------------------------------------------
__global__ __launch_bounds__(1024) void k_scanmeta(
    const float* __restrict__ bSum, const float* __restrict__ bSum2,
    float* __restrict__ bOff, float* __restrict__ scal)
{
    __shared__ float sc[1024];
    int t = threadIdx.x;
    float v = bSum[t];
    sc[t] = v;
    __syncthreads();
    for (int off = 1; off < 1024; off <<= 1) {
        float add = (t >= off) ? sc[t - off] : 0.0f;
        __syncthreads();
        sc[t] += add;
        __syncthreads();
    }
    float incl = sc[t];
    bOff[t] = incl - v;                  // exclusive offset
    float Z = sc[1023];
    __syncthreads();
    sc[t] = bSum2[t];
    __syncthreads();
    for (int s = 512; s > 0; s >>= 1) {
        if (t < s) sc[t] += sc[t + s];
        __syncthreads();
    }
    if (t == 0) {
        float E2   = sc[0];
        float neff = (Z * Z) / E2;
        scal[1] = Z;
        scal[2] = __logf(Z);
        scal[3] = (neff <= 0.25f * (float)PN) ? 1.0f : 0.0f;
    }
}

// ---------------------------------------------------------------------------
// K5: per-block inclusive prefix of e=exp(w-gmax), + block offset -> cdfE
// ---------------------------------------------------------------------------
__global__ __launch_bounds__(256) void k_scan(
    const float* __restrict__ w, const float* __restrict__ scal,
    const float* __restrict__ bOff, float* __restrict__ cdf)
{
    __shared__ float sc[256];
    int t = threadIdx.x, b = blockIdx.x;
    float gmax = scal[0];
    float4 v = ((const float4*)(w + (size_t)b * 1024))[t];
    float e0 = __expf(v.x - gmax), e1 = __expf(v.y - gmax);
    float e2 = __expf(v.z - gmax), e3 = __expf(v.w - gmax);
    float ts = ((e0 + e1) + (e2 + e3));
    sc[t] = ts;
    __syncthreads();
    for (int off = 1; off < 256; off <<= 1) {
        float add = (t >= off) ? sc[t - off] : 0.0f;
        __syncthreads();
        sc[t] += add;
        __syncthreads();
    }
    float ex = sc[t] - ts + bOff[b];
    float c0 = ex + e0, c1 = c0 + e1, c2 = c1 + e2, c3 = c2 + e3;
    float4 o; o.x = c0; o.y = c1; o.z = c2; o.w = c3;
    ((float4*)(cdf + (size_t)b * 1024))[t] = o;
}

// ---------------------------------------------------------------------------
// K6: conditional inverse-CDF resample + output [N,4]
// ---------------------------------------------------------------------------
__global__ __launch_bounds__(256) void k_resample(
    const float* __restrict__ xnew, const float* __restrict__ w,
    const float* __restrict__ cdf, const float* __restrict__ u,
    const float* __restrict__ scal, float* __restrict__ out)
{
    int i = blockIdx.x * 256 + threadIdx.x;
    float gmax = scal[0], Z = scal[1], logZ = scal[2], cond = scal[3];
    float ox, oy, oth, op;
    if (cond > 0.5f) {
        float target = u[i] * Z;                 // search cdfE for u*Z == cdf for u
        int lo = 0, hi = PN;                     // first idx with cdf[idx] >= target
        while (lo < hi) {
            int mid = (lo + hi) >> 1;
            if (cdf[mid] < target) lo = mid + 1; else hi = mid;
        }
        int idx = lo < (PN - 1) ? lo : (PN - 1);
        ox  = xnew[3 * idx + 0];
        oy  = xnew[3 * idx + 1];
        oth = xnew[3 * idx + 2];
        op  = LOG_UNIF;
    } else {
        ox  = xnew[3 * i + 0];
        oy  = xnew[3 * i + 1];
        oth = xnew[3 * i + 2];
        op  = w[i] - gmax - logZ;
    }
    float4 o; o.x = ox; o.y = oy; o.z = oth; o.w = op;
    ((float4*)out)[i] = o;
}

// ---------------------------------------------------------------------------
extern "C" void kernel_launch(void* const* d_in, const int* in_sizes, int n_in,
                              void* d_out, int out_size, void* d_ws, size_t ws_size,
                              hipStream_t stream)
{
    const float* x       = (const float*)d_in[0];
    const float* p       = (const float*)d_in[1];
    const float* sensors = (const float*)d_in[2];
    const float* z       = (const float*)d_in[3];
    const float* nth     = (const float*)d_in[4];
    const float* nsp     = (const float*)d_in[5];
    const float* u       = (const float*)d_in[6];
    const float* delta   = (const float*)d_in[7];
    float* out = (float*)d_out;

    float* xnew = (float*)d_ws;          // N*3
    float* w    = xnew + 3 * PN;         // N
    float* cdf  = w + PN;                // N
    float* pmax = cdf + PN;              // 4096
    float* bSum = pmax + 4096;           // 1024
    float* bSum2= bSum + 1024;           // 1024
    float* bOff = bSum2 + 1024;          // 1024
    float* scal = bOff + 1024;           // scalars

    k_predict_weight<<<PN / 256, 256, 0, stream>>>(x, p, sensors, z, nth, nsp,
                                                   delta, xnew, w, pmax);
    k_maxreduce<<<1, 1024, 0, stream>>>(pmax, scal);
    k_sums<<<PN / 1024, 256, 0, stream>>>(w, scal, bSum, bSum2);
    k_scanmeta<<<1, 1024, 0, stream>>>(bSum, bSum2, bOff, scal);
    k_scan<<<PN / 1024, 256, 0, stream>>>(w, scal, bOff, cdf);
    k_resample<<<PN / 256, 256, 0, stream>>>(xnew, w, cdf, u, scal, out);
}